// EmbModel_11166914970052
// MI455X (gfx1250) — compile-verified
//
#include <hip/hip_runtime.h>
#include <hip/hip_bf16.h>

#define HID   128
#define ND    4096
#define NPR   20000
#define NCL   1024
#define E_DD  500000
#define E_DP  1000000
#define E_CP  500000
#define BSZ   8192
#define SLOPE 0.2f

typedef float v2f __attribute__((ext_vector_type(2)));
typedef float v8f __attribute__((ext_vector_type(8)));

// ---------------------------------------------------------------- utilities

__global__ void fill_f32(float* p, float v, int n) {
    int i = blockIdx.x * blockDim.x + threadIdx.x;
    if (i < n) p[i] = v;
}

__device__ __forceinline__ void atomicMaxF32(float* addr, float val) {
    int* ai = (int*)addr;
    int old = __float_as_int(*addr);
    while (__int_as_float(old) < val) {
        int prev = atomicCAS(ai, old, __float_as_int(val));
        if (prev == old) break;
        old = prev;
    }
}

// out[r] = dot(X[r, 0:128], v)  — one wave (32 lanes) per row, float4 per lane
__global__ void rowdot128(const float* __restrict__ X, const float* __restrict__ v,
                          float* __restrict__ out, int n) {
    int t = blockIdx.x * blockDim.x + threadIdx.x;
    int w = t >> 5, lane = t & 31;
    if (w >= n) return;
    const float4* row = (const float4*)(X + (size_t)w * HID);
    const float4* vv  = (const float4*)v;
    float4 a = row[lane], b = vv[lane];
    float s = a.x * b.x + a.y * b.y + a.z * b.z + a.w * b.w;
    #pragma unroll
    for (int o = 16; o; o >>= 1) s += __shfl_down(s, o, 32);
    if (lane == 0) out[w] = s;
}

// v[k] = sum_c W[k*128+c] * a[c]   (W is 128x128 row-major) — <<<1,128>>>
__global__ void matvec128(const float* __restrict__ W, const float* __restrict__ a,
                          float* __restrict__ v) {
    int k = threadIdx.x;
    float s = 0.f;
    #pragma unroll 4
    for (int c = 0; c < HID; ++c) s += W[k * HID + c] * a[c];
    v[k] = s;
}

// out[0] = dot(a,b) over 128 — <<<1,32>>>
__global__ void dot128(const float* __restrict__ a, const float* __restrict__ b,
                       float* __restrict__ out) {
    int lane = threadIdx.x;
    const float4* A = (const float4*)a;
    const float4* B = (const float4*)b;
    float4 x = A[lane], y = B[lane];
    float s = x.x * y.x + x.y * y.y + x.z * y.z + x.w * y.w;
    #pragma unroll
    for (int o = 16; o; o >>= 1) s += __shfl_down(s, o, 32);
    if (lane == 0) out[0] = s;
}

// ---------------------------------------------------------------- WMMA GEMM
// C[M x N] = act(A[M x K] @ B[K x N] + bias),  f32 via V_WMMA_F32_16X16X4_F32.
// Block = 128 threads (4 waves). Each block: 16 rows x 64 cols (one 16x16
// N-tile per wave). A tile staged in LDS (padded stride to dodge bank clash).
__global__ void __launch_bounds__(128)
wmma_gemm_f32(const float* __restrict__ A, const float* __restrict__ Bm,
              const float* __restrict__ bias, float* __restrict__ C,
              int N, int K, int relu) {
    __shared__ float As[16 * 769];           // supports K up to 768, +1 pad
    const int KP = K + 1;
    const int m0 = blockIdx.y * 16;
    const int tid = threadIdx.x;

    for (int idx = tid; idx < 16 * K; idx += 128) {
        int r = idx / K, c = idx - r * K;
        As[r * KP + c] = A[(size_t)(m0 + r) * K + c];
    }
    __syncthreads();

    const int lane = tid & 31;
    const int wv   = tid >> 5;
    const int n0   = blockIdx.x * 64 + wv * 16;
    const int mn   = lane & 15;              // M index for A frag, N index for B frag
    const int kk   = (lane >> 4) << 1;       // 0 for lanes 0-15, 2 for lanes 16-31

    v8f acc = {};
    for (int k = 0; k < K; k += 4) {
        v2f a, b;
        a.x = As[mn * KP + k + kk];
        a.y = As[mn * KP + k + kk + 1];
        const float* bp = Bm + (size_t)(k + kk) * N + n0 + mn;
        b.x = bp[0];
        b.y = bp[N];
        acc = __builtin_amdgcn_wmma_f32_16x16x4_f32(false, a, false, b,
                                                    (short)0, acc, false, false);
    }

    const int col   = n0 + mn;
    const int rbase = m0 + ((lane >> 4) << 3);   // +8 for upper half-wave
    const float bc  = bias ? bias[col] : 0.f;
    #pragma unroll
    for (int j = 0; j < 8; ++j) {
        float v = acc[j] + bc;
        if (relu) v = fmaxf(v, 0.f);
        C[(size_t)(rbase + j) * N + col] = v;
    }
}

// ---------------------------------------------------------------- GAT pieces

__global__ void deg_accum(const int* __restrict__ dst, const float* __restrict__ attr,
                          float* deg, float* sattr, int E) {
    int i = blockIdx.x * blockDim.x + threadIdx.x;
    if (i >= E) return;
    int d = dst[i];
    atomicAdd(&deg[d], 1.0f);
    atomicAdd(&sattr[d], attr[i]);
}

__global__ void meanize(float* sattr, const float* deg, int n) {
    int i = blockIdx.x * blockDim.x + threadIdx.x;
    if (i < n) sattr[i] = sattr[i] / fmaxf(deg[i], 1.0f);
}

// phase 1: emax[d] = max(e); phase 2: den[d] += exp(e - emax[d]).
// Edge i>=E is a self-loop (src=dst=i-E, attr=meanattr).
__global__ void edge_softmax_pass(const int* __restrict__ src, const int* __restrict__ dst,
                                  const float* __restrict__ attr, const float* __restrict__ meanattr,
                                  const float* __restrict__ cedge,
                                  const float* __restrict__ sA, const float* __restrict__ sD,
                                  float* emax, float* den, int E, int SL, int phase) {
    int i = blockIdx.x * blockDim.x + threadIdx.x;
    if (i >= E + SL) return;
    int s, d; float at = 0.f;
    if (i < E) { s = src[i]; d = dst[i]; if (attr) at = attr[i]; }
    else       { s = d = i - E; at = meanattr[s]; }
    float e = sA[s] + sD[d];
    if (cedge) e += cedge[0] * at;
    e = e > 0.f ? e : SLOPE * e;
    if (phase == 1) atomicMaxF32(&emax[d], e);
    else            atomicAdd(&den[d], __expf(e - emax[d]));
}

// one wave per edge: acc[dst] += alpha * HS[src]  (128 channels, float4/lane)
__global__ void edge_scatter(const int* __restrict__ src, const int* __restrict__ dst,
                             const float* __restrict__ attr, const float* __restrict__ meanattr,
                             const float* __restrict__ cedge,
                             const float* __restrict__ sA, const float* __restrict__ sD,
                             const float* __restrict__ emax, const float* __restrict__ den,
                             const float* __restrict__ HS, float* __restrict__ acc,
                             int E, int SL) {
    int t = blockIdx.x * blockDim.x + threadIdx.x;
    int w = t >> 5, lane = t & 31;
    if (w >= E + SL) return;
    int s, d; float at = 0.f;
    if (w < E) { s = src[w]; d = dst[w]; if (attr) at = attr[w]; }
    else       { s = d = w - E; at = meanattr[s]; }
    float e = sA[s] + sD[d];
    if (cedge) e += cedge[0] * at;
    e = e > 0.f ? e : SLOPE * e;
    float alpha = __expf(e - emax[d]) / (den[d] + 1e-16f);
    float4 h = ((const float4*)(HS + (size_t)s * HID))[lane];
    float* arow = acc + (size_t)d * HID + lane * 4;
    atomicAdd(arow + 0, alpha * h.x);
    atomicAdd(arow + 1, alpha * h.y);
    atomicAdd(arow + 2, alpha * h.z);
    atomicAdd(arow + 3, alpha * h.w);
}

// relu(acc + b1 [+ b2]) then L2-normalize each 128-row, in place. Wave/row.
__global__ void node_finalize(float* acc, const float* __restrict__ b1,
                              const float* __restrict__ b2, int n) {
    int t = blockIdx.x * blockDim.x + threadIdx.x;
    int w = t >> 5, lane = t & 31;
    if (w >= n) return;
    float4* row = (float4*)(acc + (size_t)w * HID);
    float4 x = row[lane];
    float4 a = ((const float4*)b1)[lane];
    x.x += a.x; x.y += a.y; x.z += a.z; x.w += a.w;
    if (b2) {
        float4 c = ((const float4*)b2)[lane];
        x.x += c.x; x.y += c.y; x.z += c.z; x.w += c.w;
    }
    x.x = fmaxf(x.x, 0.f); x.y = fmaxf(x.y, 0.f);
    x.z = fmaxf(x.z, 0.f); x.w = fmaxf(x.w, 0.f);
    float ss = x.x * x.x + x.y * x.y + x.z * x.z + x.w * x.w;
    #pragma unroll
    for (int o = 16; o; o >>= 1) ss += __shfl_xor(ss, o, 32);
    float sc = 1.0f / fmaxf(sqrtf(ss), 1e-12f);
    x.x *= sc; x.y *= sc; x.z *= sc; x.w *= sc;
    row[lane] = x;
}

__global__ void build_hid(const float* __restrict__ drugN, const float* __restrict__ cellN,
                          const int* __restrict__ d1, const int* __restrict__ d2,
                          const int* __restrict__ cc, float* __restrict__ hid) {
    int t = blockIdx.x * blockDim.x + threadIdx.x;
    if (t >= BSZ * 384) return;
    int b = t / 384, c = t - b * 384;
    float v;
    if (c < 128)      v = drugN[(size_t)d1[b] * HID + c];
    else if (c < 256) v = drugN[(size_t)d2[b] * HID + (c - 128)];
    else              v = cellN[(size_t)cc[b] * HID + (c - 256)];
    hid[t] = v;
}

// out[b, j] = dot(h2[b, 0:256], W3[:, j]) + b3[j]
__global__ void final_out(const float* __restrict__ h2, const float* __restrict__ W3,
                          const float* __restrict__ b3, float* __restrict__ out) {
    int t = blockIdx.x * blockDim.x + threadIdx.x;
    if (t >= BSZ * 2) return;
    int b = t >> 1, j = t & 1;
    const float* r = h2 + (size_t)b * 256;
    float s = b3[j];
    #pragma unroll 4
    for (int k = 0; k < 256; ++k) s += r[k] * W3[k * 2 + j];
    out[t] = s;
}

// ---------------------------------------------------------------- launcher

static inline int cdiv(int a, int b) { return (a + b - 1) / b; }

extern "C" void kernel_launch(void* const* d_in, const int* in_sizes, int n_in,
                              void* d_out, int out_size, void* d_ws, size_t ws_size,
                              hipStream_t stream) {
    (void)in_sizes; (void)n_in; (void)out_size; (void)ws_size;

    const int*   ei_dd     = (const int*)d_in[3];
    const float* eattr_dd  = (const float*)d_in[4];
    const int*   src_dp    = (const int*)d_in[5];
    const int*   dst_dp    = (const int*)d_in[6];
    const int*   src_cp    = (const int*)d_in[7];
    const int*   dst_cp    = (const int*)d_in[8];
    const int*   drug1     = (const int*)d_in[9];
    const int*   drug2     = (const int*)d_in[10];
    const int*   cellb     = (const int*)d_in[11];
    const float* drug_emb  = (const float*)d_in[12];
    const float* prot_emb  = (const float*)d_in[13];
    const float* cell_emb  = (const float*)d_in[14];
    const float* W_dd      = (const float*)d_in[15];
    const float* a_src_dd  = (const float*)d_in[16];
    const float* a_dst_dd  = (const float*)d_in[17];
    const float* bias_dd   = (const float*)d_in[18];
    const float* W_edge_dd = (const float*)d_in[19];
    const float* a_edge_dd = (const float*)d_in[20];
    const float* W_dp      = (const float*)d_in[21];
    const float* a_src_dp  = (const float*)d_in[22];
    const float* a_dst_dp  = (const float*)d_in[23];
    const float* bias_dp   = (const float*)d_in[24];
    const float* W_cp      = (const float*)d_in[25];
    const float* a_src_cp  = (const float*)d_in[26];
    const float* a_dst_cp  = (const float*)d_in[27];
    const float* bias_cp   = (const float*)d_in[28];
    const float* W1        = (const float*)d_in[29];
    const float* b1        = (const float*)d_in[30];
    const float* W2        = (const float*)d_in[31];
    const float* b2        = (const float*)d_in[32];
    const float* W3        = (const float*)d_in[33];
    const float* b3        = (const float*)d_in[34];

    const int* sdd = ei_dd;            // row 0: src
    const int* ddd = ei_dd + E_DD;     // row 1: dst

    // --- scratch layout (floats, 256-aligned chunks)
    float* w = (float*)d_ws;
    size_t off = 0;
    auto alloc = [&](size_t n) { float* p = w + off; off += (n + 255) & ~size_t(255); return p; };
    float* HS_dd   = alloc((size_t)ND  * HID);
    float* HS_dp   = alloc((size_t)NPR * HID);
    float* HS_cp   = alloc((size_t)NPR * HID);
    float* sA_dd   = alloc(ND);
    float* sD_dd   = alloc(ND);
    float* sA_dp   = alloc(NPR);
    float* sD_dp   = alloc(ND);
    float* sA_cp   = alloc(NPR);
    float* sD_cp   = alloc(NCL);
    float* v_dp    = alloc(HID);
    float* v_cp    = alloc(HID);
    float* cedge   = alloc(1);
    float* deg     = alloc(ND);
    float* meanat  = alloc(ND);        // sum(attr) then mean
    float* emax_dd = alloc(ND);
    float* den_dd  = alloc(ND);
    float* emax_dp = alloc(ND);
    float* den_dp  = alloc(ND);
    float* emax_cp = alloc(NCL);
    float* den_cp  = alloc(NCL);
    float* acc_dr  = alloc((size_t)ND  * HID);
    float* acc_cl  = alloc((size_t)NCL * HID);
    float* hid     = alloc((size_t)BSZ * 384);
    float* h1      = alloc((size_t)BSZ * 768);
    float* h2      = alloc((size_t)BSZ * 256);

    // --- init accumulators
    fill_f32<<<cdiv(ND, 256), 256, 0, stream>>>(deg,     0.f,   ND);
    fill_f32<<<cdiv(ND, 256), 256, 0, stream>>>(meanat,  0.f,   ND);
    fill_f32<<<cdiv(ND, 256), 256, 0, stream>>>(emax_dd, -1e30f, ND);
    fill_f32<<<cdiv(ND, 256), 256, 0, stream>>>(den_dd,  0.f,   ND);
    fill_f32<<<cdiv(ND, 256), 256, 0, stream>>>(emax_dp, -1e30f, ND);
    fill_f32<<<cdiv(ND, 256), 256, 0, stream>>>(den_dp,  0.f,   ND);
    fill_f32<<<cdiv(NCL, 256), 256, 0, stream>>>(emax_cp, -1e30f, NCL);
    fill_f32<<<cdiv(NCL, 256), 256, 0, stream>>>(den_cp,  0.f,   NCL);
    fill_f32<<<cdiv(ND * HID, 256), 256, 0, stream>>>(acc_dr, 0.f, ND * HID);
    fill_f32<<<cdiv(NCL * HID, 256), 256, 0, stream>>>(acc_cl, 0.f, NCL * HID);

    // --- source-side projections HS = X @ W (WMMA f32)
    wmma_gemm_f32<<<dim3(2, ND / 16),  128, 0, stream>>>(drug_emb, W_dd, nullptr, HS_dd, HID, HID, 0);
    wmma_gemm_f32<<<dim3(2, NPR / 16), 128, 0, stream>>>(prot_emb, W_dp, nullptr, HS_dp, HID, HID, 0);
    wmma_gemm_f32<<<dim3(2, NPR / 16), 128, 0, stream>>>(prot_emb, W_cp, nullptr, HS_cp, HID, HID, 0);

    // --- attention scalar tables
    matvec128<<<1, 128, 0, stream>>>(W_dp, a_dst_dp, v_dp);
    matvec128<<<1, 128, 0, stream>>>(W_cp, a_dst_cp, v_cp);
    dot128<<<1, 32, 0, stream>>>(W_edge_dd, a_edge_dd, cedge);

    rowdot128<<<cdiv(ND * 32, 256),  256, 0, stream>>>(HS_dd, a_src_dd, sA_dd, ND);
    rowdot128<<<cdiv(ND * 32, 256),  256, 0, stream>>>(HS_dd, a_dst_dd, sD_dd, ND);
    rowdot128<<<cdiv(NPR * 32, 256), 256, 0, stream>>>(HS_dp, a_src_dp, sA_dp, NPR);
    rowdot128<<<cdiv(NPR * 32, 256), 256, 0, stream>>>(HS_cp, a_src_cp, sA_cp, NPR);
    rowdot128<<<cdiv(ND * 32, 256),  256, 0, stream>>>(drug_emb, v_dp, sD_dp, ND);
    rowdot128<<<cdiv(NCL * 32, 256), 256, 0, stream>>>(cell_emb, v_cp, sD_cp, NCL);

    // --- self-loop edge-attr mean (fill_value='mean')
    deg_accum<<<cdiv(E_DD, 256), 256, 0, stream>>>(ddd, eattr_dd, deg, meanat, E_DD);
    meanize<<<cdiv(ND, 256), 256, 0, stream>>>(meanat, deg, ND);

    // --- GAT d-d (E_DD edges + ND self-loops)
    const int Edd = E_DD + ND;
    edge_softmax_pass<<<cdiv(Edd, 256), 256, 0, stream>>>(sdd, ddd, eattr_dd, meanat, cedge,
                                                          sA_dd, sD_dd, emax_dd, den_dd, E_DD, ND, 1);
    edge_softmax_pass<<<cdiv(Edd, 256), 256, 0, stream>>>(sdd, ddd, eattr_dd, meanat, cedge,
                                                          sA_dd, sD_dd, emax_dd, den_dd, E_DD, ND, 2);
    edge_scatter<<<cdiv(Edd * 32, 256), 256, 0, stream>>>(sdd, ddd, eattr_dd, meanat, cedge,
                                                          sA_dd, sD_dd, emax_dd, den_dd,
                                                          HS_dd, acc_dr, E_DD, ND);

    // --- GAT protein -> drug
    edge_softmax_pass<<<cdiv(E_DP, 256), 256, 0, stream>>>(src_dp, dst_dp, nullptr, nullptr, nullptr,
                                                           sA_dp, sD_dp, emax_dp, den_dp, E_DP, 0, 1);
    edge_softmax_pass<<<cdiv(E_DP, 256), 256, 0, stream>>>(src_dp, dst_dp, nullptr, nullptr, nullptr,
                                                           sA_dp, sD_dp, emax_dp, den_dp, E_DP, 0, 2);
    edge_scatter<<<cdiv(E_DP * 32, 256), 256, 0, stream>>>(src_dp, dst_dp, nullptr, nullptr, nullptr,
                                                           sA_dp, sD_dp, emax_dp, den_dp,
                                                           HS_dp, acc_dr, E_DP, 0);

    // --- GAT protein -> cell
    edge_softmax_pass<<<cdiv(E_CP, 256), 256, 0, stream>>>(src_cp, dst_cp, nullptr, nullptr, nullptr,
                                                           sA_cp, sD_cp, emax_cp, den_cp, E_CP, 0, 1);
    edge_softmax_pass<<<cdiv(E_CP, 256), 256, 0, stream>>>(src_cp, dst_cp, nullptr, nullptr, nullptr,
                                                           sA_cp, sD_cp, emax_cp, den_cp, E_CP, 0, 2);
    edge_scatter<<<cdiv(E_CP * 32, 256), 256, 0, stream>>>(src_cp, dst_cp, nullptr, nullptr, nullptr,
                                                           sA_cp, sD_cp, emax_cp, den_cp,
                                                           HS_cp, acc_cl, E_CP, 0);

    // --- relu(sum + biases) then L2 normalize (in place)
    node_finalize<<<cdiv(ND * 32, 256),  256, 0, stream>>>(acc_dr, bias_dd, bias_dp, ND);
    node_finalize<<<cdiv(NCL * 32, 256), 256, 0, stream>>>(acc_cl, bias_cp, nullptr, NCL);

    // --- classifier
    build_hid<<<cdiv(BSZ * 384, 256), 256, 0, stream>>>(acc_dr, acc_cl, drug1, drug2, cellb, hid);
    wmma_gemm_f32<<<dim3(768 / 64, BSZ / 16), 128, 0, stream>>>(hid, W1, b1, h1, 768, 384, 1);
    wmma_gemm_f32<<<dim3(256 / 64, BSZ / 16), 128, 0, stream>>>(h1,  W2, b2, h2, 256, 768, 1);
    final_out<<<cdiv(BSZ * 2, 256), 256, 0, stream>>>(h2, W3, b3, (float*)d_out);
}